// DCT_18769007084406
// MI455X (gfx1250) — compile-verified
//
#include <hip/hip_runtime.h>

typedef float v2f __attribute__((ext_vector_type(2)));
typedef float v8f __attribute__((ext_vector_type(8)));

#define HH 512
#define WW 512
#define NB 64          // 8x8 blocks per image row/col
#define LSTR 68        // LDS row stride in words (64 + 4 pad)

__global__ __launch_bounds__(256) void dct_fused_kernel(
    const float* __restrict__ x,      // [32,3,512,512]
    const float* __restrict__ max_,   // [32,192,1,1]
    const float* __restrict__ min_,   // [32,192,1,1]
    const float* __restrict__ ycw,    // [3,3]
    const float* __restrict__ dctw,   // [192,1,8,8] (3 identical 64-filter copies)
    float* __restrict__ out)          // [32,192,64,64]
{
    __shared__ float Xs[3][NB][LSTR]; // [ycbcr ch][block col][pixel p=i*8+j]
    __shared__ float Asc[192];        // per-channel scale  d^-32
    __shared__ float Boff[192];       // per-channel offset -min*(1-d^-32)/(d-1)

    const int oh  = blockIdx.x;       // block row 0..63
    const int b   = blockIdx.y;       // batch 0..31
    const int tid = threadIdx.x;

    // ---------------- Stage 0: per-channel normalization constants ----------------
    // (exact IEEE divisions, but only 2 per channel per workgroup; overlaps stage 1)
    if (tid < 192) {
        const int mmidx = b * 192 + tid;
        const float mn = min_[mmidx];
        const float mx = max_[mmidx];
        const float d  = mx - mn + 1e-6f;
        float A = 1.0f / d;                                   // d^-1
        A = A * A; A = A * A; A = A * A; A = A * A; A = A * A; // d^-32
        Asc[tid]  = A;
        Boff[tid] = -mn * (1.0f - A) / (d - 1.0f);
    }

    // ---------------- Stage 1: RGB -> YCbCr into LDS ----------------
    const float w00 = ycw[0], w01 = ycw[1], w02 = ycw[2];
    const float w10 = ycw[3], w11 = ycw[4], w12 = ycw[5];
    const float w20 = ycw[6], w21 = ycw[7], w22 = ycw[8];

    const float* xb = x + (size_t)b * 3 * HH * WW + (size_t)(8 * oh) * WW;
    #pragma unroll
    for (int it = 0; it < 4; ++it) {
        const int q   = tid + 256 * it;   // 0..1023 float4-chunks
        const int i   = q >> 7;           // row in block 0..7
        const int cc  = q & 127;          // col chunk 0..127
        const int col = cc * 4;
        const float4 r4 = *(const float4*)(xb + 0 * HH * WW + i * WW + col);
        const float4 g4 = *(const float4*)(xb + 1 * HH * WW + i * WW + col);
        const float4 b4 = *(const float4*)(xb + 2 * HH * WW + i * WW + col);
        const int blk = cc >> 1;
        const int p   = i * 8 + (cc & 1) * 4;
        float4 t;
        t.x = w00*r4.x + w01*g4.x + w02*b4.x;
        t.y = w00*r4.y + w01*g4.y + w02*b4.y;
        t.z = w00*r4.z + w01*g4.z + w02*b4.z;
        t.w = w00*r4.w + w01*g4.w + w02*b4.w;
        *(float4*)&Xs[0][blk][p] = t;
        t.x = w10*r4.x + w11*g4.x + w12*b4.x;
        t.y = w10*r4.y + w11*g4.y + w12*b4.y;
        t.z = w10*r4.z + w11*g4.z + w12*b4.z;
        t.w = w10*r4.w + w11*g4.w + w12*b4.w;
        *(float4*)&Xs[1][blk][p] = t;
        t.x = w20*r4.x + w21*g4.x + w22*b4.x;
        t.y = w20*r4.y + w21*g4.y + w22*b4.y;
        t.z = w20*r4.z + w21*g4.z + w22*b4.z;
        t.w = w20*r4.w + w21*g4.w + w22*b4.w;
        *(float4*)&Xs[2][blk][p] = t;
    }
    __syncthreads();

    // ---------------- Stage 2: WMMA GEMM Out = W(64x64) * X(64x64) ----------------
    const int lane = tid & 31;     // wave32
    const int wave = tid >> 5;     // 0..7
    const int lrow = lane & 15;    // M / N index within fragment
    const int hi   = lane >> 4;    // K-pair select (A/B), +8 M offset (D)

    #pragma unroll
    for (int jj = 0; jj < 3; ++jj) {
        const int j   = wave + 8 * jj;   // 0..23
        const int c   = j >> 3;          // ycbcr channel
        const int rem = j & 7;
        const int m   = rem >> 1;        // M tile (16 coeffs)
        const int nh  = rem & 1;         // N half (32 blocks)

        // A fragments: W row (m*16+lrow), K pairs {4k+2hi, 4k+2hi+1}
        const float* wrowp = dctw + (size_t)(m * 16 + lrow) * 64 + 2 * hi;
        v2f a[16];
        #pragma unroll
        for (int k = 0; k < 16; ++k)
            a[k] = *(const v2f*)(wrowp + 4 * k);

        #pragma unroll
        for (int t = 0; t < 2; ++t) {
            const int nt   = nh * 2 + t;
            const int ncol = nt * 16 + lrow;        // block column (ow)
            const float* bcol = &Xs[c][ncol][2 * hi];

            v8f acc = {0.f, 0.f, 0.f, 0.f, 0.f, 0.f, 0.f, 0.f};
            #pragma unroll
            for (int k = 0; k < 16; ++k) {
                v2f bf = *(const v2f*)(bcol + 4 * k);   // single ds_load_b64 pair
                acc = __builtin_amdgcn_wmma_f32_16x16x4_f32(
                    /*neg_a=*/false, a[k], /*neg_b=*/false, bf,
                    /*c_mod=*/(short)0, acc, /*reuse_a=*/false, /*reuse_b=*/false);
            }

            // ---- Stage 3: folded 32x normalization (precomputed) + store ----
            #pragma unroll
            for (int r = 0; r < 8; ++r) {
                const int kcoef = m * 16 + 8 * hi + r;   // D layout: VGPR r, hi half = +8
                const int o     = c * 64 + kcoef;
                const float v   = acc[r] * Asc[o] + Boff[o];
                out[(((size_t)b * 192 + o) * NB + oh) * NB + ncol] = v;
            }
        }
    }
}

extern "C" void kernel_launch(void* const* d_in, const int* in_sizes, int n_in,
                              void* d_out, int out_size, void* d_ws, size_t ws_size,
                              hipStream_t stream) {
    const float* x    = (const float*)d_in[0];
    const float* max_ = (const float*)d_in[1];
    const float* min_ = (const float*)d_in[2];
    const float* ycw  = (const float*)d_in[3];
    const float* dctw = (const float*)d_in[4];
    float* out = (float*)d_out;
    (void)in_sizes; (void)n_in; (void)out_size; (void)d_ws; (void)ws_size;

    dim3 grid(NB, 32);   // (block-row oh, batch b)
    dct_fused_kernel<<<grid, 256, 0, stream>>>(x, max_, min_, ycw, dctw, out);
}